// RoIPointPool3d_23845658427905
// MI455X (gfx1250) — compile-verified
//
#include <hip/hip_runtime.h>

// Problem constants (match the reference file).
constexpr int Bb = 4;
constexpr int Nn = 16384;
constexpr int Mm = 128;
constexpr int Cc = 128;
constexpr int Ss = 512;
constexpr int OUTC = 3 + Cc;        // 131 floats per pooled row
constexpr int BLK = 512;            // 16 wave32 waves per block
constexpr int NWAVE = BLK / 32;     // 16
constexpr int NCHUNK = Nn / BLK;    // 32 chunks of 512 points
constexpr int CHUNK_BYTES = BLK * 12;         // 6144 B of xyz per chunk
constexpr int CHUNK_XFERS = CHUNK_BYTES / 16; // 384 b128 transfers

// Low 32 bits of a generic pointer to LDS == LDS byte address (CDNA5 aperture
// places SHARED_BASE in addr[63:32]; the low word is the LDS offset).
__device__ __forceinline__ unsigned lds_lo32(const void* p) {
  return (unsigned)(unsigned long long)p;
}

// gfx1250 async copy: 16 B global -> LDS per active lane, tracked by ASYNCcnt.
__device__ __forceinline__ void async_ld16(unsigned lds_byte, const void* gaddr) {
  asm volatile("global_load_async_to_lds_b128 %0, %1, off"
               : : "v"(lds_byte), "v"(gaddr) : "memory");
}

#define WAIT_ASYNCCNT(n) asm volatile("s_wait_asynccnt %0" : : "i"(n) : "memory")

__global__ __launch_bounds__(BLK)
void roipool3d_kernel(const float* __restrict__ points,       // (B,N,3)
                      const float* __restrict__ feats,        // (B,N,C)
                      const float* __restrict__ boxes,        // (B,M,7)
                      float* __restrict__ out,                // (B,M,S,131)
                      float* __restrict__ flags) {            // (B,M) as float
  __shared__ __align__(16) float s_pts[2][BLK * 3];  // double-buffered xyz chunks
  __shared__ int s_table[Ss];                        // first-come in-box indices
  __shared__ int s_wsum[NWAVE];                      // per-wave mask popcounts

  const int m    = blockIdx.x;
  const int b    = blockIdx.y;
  const int tid  = threadIdx.x;
  const int lane = tid & 31;
  const int wave = tid >> 5;

  // ---- box parameters (uniform -> scalar loads) -------------------------
  const float* bx = boxes + ((size_t)b * Mm + m) * 7;
  const float cx = bx[0], cy = bx[1];
  const float dz = bx[5];
  const float cz = bx[2] + 0.5f * dz;               // bottom-center -> center
  const float hx = 0.5f * bx[3];
  const float hy = 0.5f * bx[4];
  const float hz = 0.5f * dz;
  const float rz = bx[6];
  // rz in [-pi, pi]: hardware v_cos/v_sin are exact enough; avoids the
  // Payne-Hanek big-argument SALU blob of libm cosf/sinf.
  const float cosa = __cosf(-rz);
  const float sina = __sinf(-rz);

  const char* gbase = (const char*)(points + (size_t)b * Nn * 3);

  auto stage = [&](int ci, int buf) {
    if (tid < CHUNK_XFERS) {
      unsigned l = lds_lo32(&s_pts[buf][0]) + tid * 16;
      const char* g = gbase + (size_t)ci * CHUNK_BYTES + tid * 16;
      async_ld16(l, g);
    }
  };

  // ---- phase 1: stable compaction of in-box point indices ---------------
  stage(0, 0);
  int base = 0;  // running in-box count; identical in every thread
  for (int ci = 0; ci < NCHUNK; ++ci) {
    const int buf = ci & 1;
    if (ci + 1 < NCHUNK) {
      stage(ci + 1, buf ^ 1);
      WAIT_ASYNCCNT(1);   // oldest (current chunk) complete; next may be in flight
    } else {
      WAIT_ASYNCCNT(0);
    }
    __syncthreads();      // (A) staged chunk visible to all waves

    const int p = ci * BLK + tid;
    const float x = s_pts[buf][tid * 3 + 0];
    const float y = s_pts[buf][tid * 3 + 1];
    const float z = s_pts[buf][tid * 3 + 2];
    const float sx = x - cx, sy = y - cy;
    const float lx = sx * cosa - sy * sina;
    const float ly = sx * sina + sy * cosa;
    const bool inb = (fabsf(z - cz) <= hz) &&
                     (lx > -hx) && (lx < hx) &&
                     (ly > -hy) && (ly < hy);

    const unsigned mb = (unsigned)__ballot(inb);   // wave32: low 32 bits
    if (lane == 0) s_wsum[wave] = __popc(mb);
    __syncthreads();      // (B) wave sums visible

    int wpre = 0, tot = 0;
    for (int w = 0; w < NWAVE; ++w) {
      const int v = s_wsum[w];
      if (w < wave) wpre += v;
      tot += v;
    }
    const int pos = base + wpre + __popc(mb & ((1u << lane) - 1u));
    if (inb && pos < Ss) s_table[pos] = p;
    base += tot;
    // Once we hold S indices, k % cnt == k for all k < S: count past S is moot.
    if (base >= Ss) break;
  }
  __syncthreads();        // (C) table complete

  // ---- phase 2: gather + write ------------------------------------------
  const int cnt = base;
  float* boxout = out + ((size_t)b * Mm + m) * (size_t)Ss * OUTC;

  if (cnt > 0) {
    const float* pb = points + (size_t)b * Nn * 3;
    const float* fb = feats + (size_t)b * Nn * Cc;
    for (int s = wave; s < Ss; s += NWAVE) {
      const int idx = s_table[s % cnt];            // LDS broadcast
      float* row = boxout + (size_t)s * OUTC;
      if (lane < 3)
        __builtin_nontemporal_store(pb[(size_t)idx * 3 + lane], row + lane);
      // one aligned b128 per lane covers the 512 B feature row of this sample;
      // loads stay regular-temporal (rows are re-read across the 128 boxes of
      // a batch and fit in L2), stores are NT (output is write-once).
      const float4 v = ((const float4*)(fb + (size_t)idx * Cc))[lane];
      float* r = row + 3 + lane * 4;               // rows are 4 B-aligned only
      __builtin_nontemporal_store(v.x, r + 0);
      __builtin_nontemporal_store(v.y, r + 1);
      __builtin_nontemporal_store(v.z, r + 2);
      __builtin_nontemporal_store(v.w, r + 3);
    }
  } else {
    for (int i = tid; i < Ss * OUTC; i += BLK)
      __builtin_nontemporal_store(0.0f, boxout + i);
  }
  if (tid == 0) flags[(size_t)b * Mm + m] = (cnt == 0) ? 1.0f : 0.0f;
}

extern "C" void kernel_launch(void* const* d_in, const int* in_sizes, int n_in,
                              void* d_out, int out_size, void* d_ws, size_t ws_size,
                              hipStream_t stream) {
  (void)in_sizes; (void)n_in; (void)out_size; (void)d_ws; (void)ws_size;
  const float* points = (const float*)d_in[0];   // (B,N,3)  f32
  const float* feats  = (const float*)d_in[1];   // (B,N,C)  f32
  const float* boxes  = (const float*)d_in[2];   // (B,M,7)  f32
  float* out   = (float*)d_out;                                  // pooled first
  float* flags = out + (size_t)Bb * Mm * Ss * OUTC;              // then flags

  dim3 grid(Mm, Bb);
  roipool3d_kernel<<<grid, BLK, 0, stream>>>(points, feats, boxes, out, flags);
}